// SupConLoss_37838661878011
// MI455X (gfx1250) — compile-verified
//
#include <hip/hip_runtime.h>
#include <math.h>

#define N_ROWS 65536
#define DIMS   1024
#define NCLS   64
#define KSEL   5
#define MSEL   (NCLS * KSEL)   // 320

typedef __attribute__((ext_vector_type(2))) float v2f;
typedef __attribute__((ext_vector_type(8))) float v8f;

// ---------------------------------------------------------------- init counts
__global__ void supcon_init_counts(int* __restrict__ counts) {
    if (threadIdx.x < NCLS) counts[threadIdx.x] = 0;
}

// --------------------------------------------------- softmax conf + argmax
// one wave32 per row; lane holds logits[row][lane] and logits[row][lane+32]
__global__ void supcon_conf_kernel(const float* __restrict__ logits,
                                   float* __restrict__ conf,
                                   int*   __restrict__ lab,
                                   int*   __restrict__ counts) {
    int wid  = (blockIdx.x * blockDim.x + threadIdx.x) >> 5;
    int lane = threadIdx.x & 31;
    if (wid >= N_ROWS) return;
    const float* row = logits + (size_t)wid * NCLS;
    float v0 = row[lane];
    float v1 = row[lane + 32];
    float bv; int bi;
    if (v0 >= v1) { bv = v0; bi = lane; } else { bv = v1; bi = lane + 32; }
    // butterfly max with first-index tie-break (matches jnp.argmax)
    for (int off = 16; off > 0; off >>= 1) {
        float ov = __shfl_xor(bv, off, 32);
        int   oi = __shfl_xor(bi, off, 32);
        if (ov > bv || (ov == bv && oi < bi)) { bv = ov; bi = oi; }
    }
    float e = expf(v0 - bv) + expf(v1 - bv);
    for (int off = 16; off > 0; off >>= 1) e += __shfl_xor(e, off, 32);
    if (lane == 0) {
        conf[wid] = 1.0f / e;          // max prob = exp(0)/sum
        lab[wid]  = bi;
        atomicAdd(&counts[bi], 1);
    }
}

// ---------------------------------------------------- per-class top-5 select
// one block per class; 5 sequential argmax passes (conf/lab are L2-resident)
__global__ void supcon_topk_kernel(const float* __restrict__ conf,
                                   const int*   __restrict__ lab,
                                   const int*   __restrict__ counts,
                                   int* __restrict__ top_idx,
                                   int* __restrict__ slot_valid) {
    __shared__ float s_val[256];
    __shared__ int   s_idx[256];
    __shared__ int   s_sel[KSEL];
    const int c = blockIdx.x, tid = threadIdx.x;
    for (int k = 0; k < KSEL; ++k) {
        float bv = -INFINITY; int bi = 0x7fffffff;
        for (int i = tid; i < N_ROWS; i += 256) {
            bool skip = false;
            for (int s = 0; s < k; ++s) if (s_sel[s] == i) skip = true;
            if (skip) continue;
            // non-members act as -inf (matches masked_conf in reference)
            float v = (lab[i] == c) ? conf[i] : -INFINITY;
            if (v > bv || (v == bv && i < bi)) { bv = v; bi = i; }
        }
        s_val[tid] = bv; s_idx[tid] = bi;
        __syncthreads();
        for (int s = 128; s > 0; s >>= 1) {
            if (tid < s) {
                float ov = s_val[tid + s]; int oi = s_idx[tid + s];
                if (ov > s_val[tid] || (ov == s_val[tid] && oi < s_idx[tid])) {
                    s_val[tid] = ov; s_idx[tid] = oi;
                }
            }
            __syncthreads();
        }
        if (tid == 0) { s_sel[k] = s_idx[0]; top_idx[c * KSEL + k] = s_idx[0]; }
        __syncthreads();
    }
    if (tid == 0) {
        int cnt  = counts[c];
        int kf   = (int)((float)cnt * 0.1f);          // trunc, as astype(int32)
        int kpc  = kf  < KSEL ? kf  : KSEL;           // min(.., TOPK_CAP)
        int kmin = cnt < KSEL ? cnt : KSEL;           // min(MIN_SAMPLES, counts)
        if (kpc < kmin) kpc = kmin;                   // max(..)
        for (int j = 0; j < KSEL; ++j) slot_valid[c * KSEL + j] = (j < kpc) ? 1 : 0;
    }
}

// ------------------------------------------- gather + L2-normalize 320 rows
__global__ void supcon_gather_norm(const float* __restrict__ feat,
                                   const int*   __restrict__ top_idx,
                                   const int*   __restrict__ lab,
                                   float* __restrict__ fn,
                                   int*   __restrict__ labm) {
    __shared__ float s_red[256];
    __shared__ float s_inv;
    const int m = blockIdx.x, tid = threadIdx.x;
    const int src = top_idx[m];
    const float* row = feat + (size_t)src * DIMS;
    float ss = 0.f;
    for (int d = tid; d < DIMS; d += 256) { float v = row[d]; ss += v * v; }
    s_red[tid] = ss; __syncthreads();
    for (int s = 128; s > 0; s >>= 1) { if (tid < s) s_red[tid] += s_red[tid + s]; __syncthreads(); }
    if (tid == 0) {
        float nrm = fmaxf(sqrtf(s_red[0]), 1e-12f);
        s_inv = 1.0f / nrm;
        labm[m] = lab[src];
    }
    __syncthreads();
    const float inv = s_inv;
    for (int d = tid; d < DIMS; d += 256) fn[(size_t)m * DIMS + d] = row[d] * inv;
}

// --------------------------------------------------- sim = f @ f^T via WMMA
// one wave per 16x16 tile; K loop in steps of 4 with V_WMMA_F32_16X16X4_F32.
// A-tile lane layout == B-tile lane layout (B[k][n] = f[n][k]).
__global__ void supcon_gemm_wmma(const float* __restrict__ fn,
                                 float* __restrict__ sim) {
    const int wid  = (blockIdx.x << 2) + (threadIdx.x >> 5);   // 400 tiles
    const int lane = threadIdx.x & 31;
    const int tm = wid / 20, tn = wid % 20;
    const int half = lane >> 4;          // 0: K={0,1}, 1: K={2,3}
    const int l16  = lane & 15;
    const float* pa = fn + (size_t)(tm * 16 + l16) * DIMS + half * 2;
    const float* pb = fn + (size_t)(tn * 16 + l16) * DIMS + half * 2;
    v8f acc = {0.f, 0.f, 0.f, 0.f, 0.f, 0.f, 0.f, 0.f};
#pragma unroll 8
    for (int k = 0; k < DIMS; k += 4) {
        v2f a = *(const v2f*)(pa + k);
        v2f b = *(const v2f*)(pb + k);
        acc = __builtin_amdgcn_wmma_f32_16x16x4_f32(
            /*neg_a=*/false, a, /*neg_b=*/false, b,
            /*c_mod=*/(short)0, acc, /*reuse_a=*/false, /*reuse_b=*/false);
    }
    const int col = tn * 16 + l16;
#pragma unroll
    for (int r = 0; r < 8; ++r) {
        int rowm = tm * 16 + half * 8 + r;     // C/D layout: VGPR r, half-split M
        sim[(size_t)rowm * MSEL + col] = acc[r];
    }
}

// --------------------------------------------------------------- final loss
__global__ void supcon_loss_kernel(const float* __restrict__ sim,
                                   const int* __restrict__ slot_valid,
                                   const int* __restrict__ labm,
                                   float* __restrict__ out) {
    __shared__ int   s_valid[MSEL];
    __shared__ int   s_lab[MSEL];
    __shared__ float s_a[MSEL];
    __shared__ float s_b[MSEL];
    __shared__ float s_temp, s_bs;
    const int t = threadIdx.x;
    s_valid[t] = slot_valid[t];
    s_lab[t]   = labm[t];
    __syncthreads();
    if (t == 0) {
        int bsum = 0;
        for (int j = 0; j < MSEL; ++j) bsum += s_valid[j];
        s_bs   = (float)bsum;
        s_temp = 0.07f * ((float)bsum * 2.0f - 1.0f);   // TEMPERATURE*(bs*2-1)
    }
    __syncthreads();
    const float invt = 1.0f / s_temp;
    const float* row = sim + (size_t)t * MSEL;
    // row max over valid cols (diagonal included, as in reference)
    float rmax = -INFINITY;
    for (int j = 0; j < MSEL; ++j)
        if (s_valid[j]) rmax = fmaxf(rmax, row[j] * invt);
    float denom = 0.f, possum = 0.f;
    int   poscnt = 0;
    const int myl = s_lab[t], myv = s_valid[t];
    for (int j = 0; j < MSEL; ++j) {
        if (j == t || !s_valid[j]) continue;
        float ss = row[j] * invt - rmax;
        denom += expf(ss);
        if (myv && s_lab[j] == myl) { possum += ss; poscnt++; }
    }
    float contrib = 0.f, anch = 0.f;
    if (myv && poscnt > 0) {
        float logden = logf(denom + 1e-8f);
        contrib = (possum - (float)poscnt * logden) / ((float)poscnt + 1e-8f);
        anch = 1.0f;
    }
    s_a[t] = contrib; s_b[t] = anch;
    __syncthreads();
    if (t == 0) {
        float ls = 0.f, na = 0.f;
        for (int j = 0; j < MSEL; ++j) { ls += s_a[j]; na += s_b[j]; }
        float loss = -ls / fmaxf(na, 1.0f);
        if (!(s_bs >= 2.0f && na > 0.0f)) loss = 0.0f;
        out[0] = loss;
    }
}

// ---------------------------------------------------------------------------
extern "C" void kernel_launch(void* const* d_in, const int* in_sizes, int n_in,
                              void* d_out, int out_size, void* d_ws, size_t ws_size,
                              hipStream_t stream) {
    const float* features = (const float*)d_in[0];   // (N, D) f32
    const float* logits   = (const float*)d_in[1];   // (N, C) f32
    float* out = (float*)d_out;

    char* w = (char*)d_ws;
    float* fn         = (float*)w; w += sizeof(float) * (size_t)MSEL * DIMS; // 1.31 MB
    float* sim        = (float*)w; w += sizeof(float) * (size_t)MSEL * MSEL; // 410 KB
    float* conf       = (float*)w; w += sizeof(float) * N_ROWS;              // 256 KB
    int*   lab        = (int*)  w; w += sizeof(int)   * N_ROWS;              // 256 KB
    int*   counts     = (int*)  w; w += sizeof(int)   * NCLS;
    int*   top_idx    = (int*)  w; w += sizeof(int)   * MSEL;
    int*   slot_valid = (int*)  w; w += sizeof(int)   * MSEL;
    int*   labm       = (int*)  w; w += sizeof(int)   * MSEL;

    supcon_init_counts<<<1, 64, 0, stream>>>(counts);
    supcon_conf_kernel<<<N_ROWS / 8, 256, 0, stream>>>(logits, conf, lab, counts);
    supcon_topk_kernel<<<NCLS, 256, 0, stream>>>(conf, lab, counts, top_idx, slot_valid);
    supcon_gather_norm<<<MSEL, 256, 0, stream>>>(features, top_idx, lab, fn, labm);
    supcon_gemm_wmma<<<100, 128, 0, stream>>>(fn, sim);   // 400 waves = 400 tiles
    supcon_loss_kernel<<<1, MSEL, 0, stream>>>(sim, slot_valid, labm, out);
}